// GIN_classifier_29609504539440
// MI455X (gfx1250) — compile-verified
//
#include <hip/hip_runtime.h>

typedef __attribute__((ext_vector_type(2))) float v2f;
typedef __attribute__((ext_vector_type(8))) float v8f;

#define N_NODES 50000
#define N_EDGES 800000
#define IN_DIM  128
#define HID_DIM 256
#define OUT_DIM 2

// ---------------------------------------------------------------------------
// Zero a float region (grid-stride).
// ---------------------------------------------------------------------------
__global__ void zero_f32(float* __restrict__ p, int n) {
  int i = blockIdx.x * blockDim.x + threadIdx.x;
  int stride = gridDim.x * blockDim.x;
  for (; i < n; i += stride) p[i] = 0.0f;
}

// ---------------------------------------------------------------------------
// Scatter-add 128-dim rows: one wave32 per edge, each lane owns 4 dims.
// agg[dst[e], :] += x[src[e], :]
// ---------------------------------------------------------------------------
__global__ void scatter_add_128(const float* __restrict__ x,
                                const int* __restrict__ src,
                                const int* __restrict__ dst,
                                float* __restrict__ agg) {
  int gid  = blockIdx.x * blockDim.x + threadIdx.x;
  int edge = gid >> 5;
  int lane = gid & 31;
  if (edge >= N_EDGES) return;
  int s = src[edge];
  int d = dst[edge];
  const float4 v = *(const float4*)(x + (size_t)s * IN_DIM + lane * 4);
  float* o = agg + (size_t)d * IN_DIM + lane * 4;
  atomicAdd(o + 0, v.x);
  atomicAdd(o + 1, v.y);
  atomicAdd(o + 2, v.z);
  atomicAdd(o + 3, v.w);
}

// ---------------------------------------------------------------------------
// Scatter-add 256-dim rows: one wave32 per edge, each lane owns 8 dims.
// ---------------------------------------------------------------------------
__global__ void scatter_add_256(const float* __restrict__ h,
                                const int* __restrict__ src,
                                const int* __restrict__ dst,
                                float* __restrict__ agg) {
  int gid  = blockIdx.x * blockDim.x + threadIdx.x;
  int edge = gid >> 5;
  int lane = gid & 31;
  if (edge >= N_EDGES) return;
  int s = src[edge];
  int d = dst[edge];
  const float4* in = (const float4*)(h + (size_t)s * HID_DIM + lane * 8);
  float* o = agg + (size_t)d * HID_DIM + lane * 8;
  float4 v0 = in[0];
  float4 v1 = in[1];
  atomicAdd(o + 0, v0.x);
  atomicAdd(o + 1, v0.y);
  atomicAdd(o + 2, v0.z);
  atomicAdd(o + 3, v0.w);
  atomicAdd(o + 4, v1.x);
  atomicAdd(o + 5, v1.y);
  atomicAdd(o + 6, v1.z);
  atomicAdd(o + 7, v1.w);
}

// ---------------------------------------------------------------------------
// GIN layer 1 GEMM via V_WMMA_F32_16X16X4_F32:
//   h = relu(((1+eps1)*x + agg1) @ W1 + b1)
// One wave per 16x16 output tile. 50000/16 = 3125 M-tiles, 256/16 = 16 N-tiles.
// 8 waves per 256-thread block -> 6250 blocks cover all 50000 tiles exactly,
// so EXEC is always all-1s (WMMA requirement).
//
// f32 operand layouts (ISA 7.12.2, wave32):
//   A 16x4 : lane&15 = M; lanes[0:15] K={k,k+1} in VGPR{0,1}, lanes[16:31] K={k+2,k+3}
//   B 4x16 : lane&15 = N; same K half-split across lane groups
//   C 16x16: VGPR r -> M = r + (lane>=16 ? 8 : 0), N = lane&15
// ---------------------------------------------------------------------------
__global__ void __launch_bounds__(256)
gin_gemm1_wmma(const float* __restrict__ x,
               const float* __restrict__ agg1,
               const float* __restrict__ W1,
               const float* __restrict__ b1,
               const float* __restrict__ eps1p,
               float* __restrict__ h) {
  const int lane = threadIdx.x & 31;
  const int wave = threadIdx.x >> 5;
  const int tile = blockIdx.x * 8 + wave;     // 0 .. 49999
  const int mTile = tile >> 4;                // 0 .. 3124
  const int nTile = tile & 15;                // 0 .. 15

  const float scale = 1.0f + eps1p[0];

  const int idx  = lane & 15;                 // M row (A) / N col (B)
  const int koff = (lane >> 4) * 2;           // K half select

  const int m = mTile * 16 + idx;
  const int n = nTile * 16 + idx;

  const float* xrow = x    + (size_t)m * IN_DIM;
  const float* arow = agg1 + (size_t)m * IN_DIM;

  v8f c = {};
#pragma unroll 8
  for (int k = 0; k < IN_DIM; k += 4) {
    const float2 xv = *(const float2*)(xrow + k + koff);
    const float2 av = *(const float2*)(arow + k + koff);
    v2f a, b;
    a.x = fmaf(scale, xv.x, av.x);
    a.y = fmaf(scale, xv.y, av.y);
    b.x = W1[(size_t)(k + koff)     * HID_DIM + n];
    b.y = W1[(size_t)(k + koff + 1) * HID_DIM + n];
    // (neg_a, A, neg_b, B, c_mod, C, reuse_a, reuse_b)
    c = __builtin_amdgcn_wmma_f32_16x16x4_f32(false, a, false, b,
                                              (short)0, c, false, false);
  }

  // bias + relu + store: this lane's column is nTile*16+idx for all 8 VGPRs
  const float bias  = b1[nTile * 16 + idx];
  const int   mBase = mTile * 16 + (lane >> 4) * 8;
  float* hout = h + (size_t)mBase * HID_DIM + nTile * 16 + idx;
#pragma unroll
  for (int r = 0; r < 8; ++r) {
    float v = c[r] + bias;
    v = v > 0.0f ? v : 0.0f;
    hout[(size_t)r * HID_DIM] = v;
  }
}

// ---------------------------------------------------------------------------
// GIN layer 2 (skinny GEMM, OUT_DIM = 2): one thread per node.
//   out[n,:] = ((1+eps2)*h[n,:] + agg2[n,:]) @ W2 + b2
// ---------------------------------------------------------------------------
__global__ void gin_out(const float* __restrict__ h,
                        const float* __restrict__ agg2,
                        const float* __restrict__ W2,
                        const float* __restrict__ b2,
                        const float* __restrict__ eps2p,
                        float* __restrict__ out) {
  int node = blockIdx.x * blockDim.x + threadIdx.x;
  if (node >= N_NODES) return;
  const float scale = 1.0f + eps2p[0];
  const float* hr = h    + (size_t)node * HID_DIM;
  const float* ar = agg2 + (size_t)node * HID_DIM;
  float acc0 = 0.0f, acc1 = 0.0f;
#pragma unroll 8
  for (int d = 0; d < HID_DIM; ++d) {
    float z = fmaf(scale, hr[d], ar[d]);
    acc0 = fmaf(z, W2[d * 2 + 0], acc0);
    acc1 = fmaf(z, W2[d * 2 + 1], acc1);
  }
  out[node * 2 + 0] = acc0 + b2[0];
  out[node * 2 + 1] = acc1 + b2[1];
}

// ---------------------------------------------------------------------------
extern "C" void kernel_launch(void* const* d_in, const int* in_sizes, int n_in,
                              void* d_out, int out_size, void* d_ws, size_t ws_size,
                              hipStream_t stream) {
  const float* x    = (const float*)d_in[0];
  const int*   ei   = (const int*)d_in[1];   // [2, N_EDGES]: row0=src, row1=dst
  const float* W1   = (const float*)d_in[2];
  const float* b1   = (const float*)d_in[3];
  const float* W2   = (const float*)d_in[4];
  const float* b2   = (const float*)d_in[5];
  const float* eps1 = (const float*)d_in[6];
  const float* eps2 = (const float*)d_in[7];
  float* out = (float*)d_out;

  const int* src = ei;
  const int* dst = ei + N_EDGES;

  // Workspace layout (128 MB total):
  float* agg1 = (float*)d_ws;                          //  50000*128 f32
  float* h    = agg1 + (size_t)N_NODES * IN_DIM;       //  50000*256 f32
  float* agg2 = h    + (size_t)N_NODES * HID_DIM;      //  50000*256 f32

  // 1) zero accumulation buffers (every call: harness doesn't re-zero ws)
  zero_f32<<<2048, 256, 0, stream>>>(agg1, N_NODES * IN_DIM);
  zero_f32<<<2048, 256, 0, stream>>>(agg2, N_NODES * HID_DIM);

  // 2) agg1 = scatter_add(x[src] -> dst)   (one wave32 per edge)
  {
    int threads = N_EDGES * 32;
    scatter_add_128<<<threads / 256, 256, 0, stream>>>(x, src, dst, agg1);
  }

  // 3) h = relu(((1+eps1)*x + agg1) @ W1 + b1)   via v_wmma_f32_16x16x4_f32
  gin_gemm1_wmma<<<(N_NODES / 16) * 16 / 8, 256, 0, stream>>>(x, agg1, W1, b1, eps1, h);

  // 4) agg2 = scatter_add(h[src] -> dst)
  {
    int threads = N_EDGES * 32;
    scatter_add_256<<<threads / 256, 256, 0, stream>>>(h, src, dst, agg2);
  }

  // 5) out = ((1+eps2)*h + agg2) @ W2 + b2
  gin_out<<<(N_NODES + 255) / 256, 256, 0, stream>>>(h, agg2, W2, b2, eps2, out);
}